// DynamicModel_28484223107438
// MI455X (gfx1250) — compile-verified
//
#include <hip/hip_runtime.h>
#include <math.h>
#include <stdint.h>

// ---- model constants (from reference) ----
constexpr float kLF   = 1.6f;
constexpr float kLR   = 1.4f;
constexpr float kCM1  = 8000.0f;
constexpr float kCM2  = 8000.0f;
constexpr float kDT   = 0.01f;
constexpr float kG    = 9.8f;
constexpr float kInvMass = 1.0f / 1845.0f;
constexpr float kMassIz  = 1845.0f / 3500.0f;
constexpr int   kEPT  = 4;      // elements per thread

// ---------------- gfx1250 async global->LDS (ASYNCcnt) ----------------
__device__ __forceinline__ void async_b128_to_lds(unsigned lds_off, const void* gptr) {
    unsigned long long ga = (unsigned long long)(uintptr_t)gptr;
    asm volatile("global_load_async_to_lds_b128 %0, %1, off"
                 :: "v"(lds_off), "v"(ga)
                 : "memory");
}

__device__ __forceinline__ void wait_async0() {
#if __has_builtin(__builtin_amdgcn_s_wait_asynccnt)
    __builtin_amdgcn_s_wait_asynccnt(0);
#else
    asm volatile("s_wait_asynccnt 0" ::: "memory");
#endif
}

// ---------------- fast transcendentals (error << tolerance; see analysis) --
// atan(t), t in [0,1]:  t + t*s*q(s), |err| <= 2e-8  (A&S 4.4.49 minimax)
__device__ __forceinline__ float atan_poly01(float t) {
    float s = t * t;
    float q = fmaf(s, 2.8662257e-03f, -1.61657367e-02f);
    q = fmaf(s, q,  4.29096138e-02f);
    q = fmaf(s, q, -7.52896400e-02f);
    q = fmaf(s, q,  1.065626393e-01f);
    q = fmaf(s, q, -1.420889944e-01f);
    q = fmaf(s, q,  1.999355085e-01f);
    q = fmaf(s, q, -3.333314528e-01f);
    return fmaf(t * s, q, t);
}

__device__ __forceinline__ float fast_atan2f(float y, float x) {
    const float ax = fabsf(x), ay = fabsf(y);
    const float mx = fmaxf(ax, ay), mn = fminf(ax, ay);
    float t = mn * __builtin_amdgcn_rcpf(mx);   // v_rcp_f32, <=1 ulp
    t = (mx == 0.0f) ? 0.0f : t;                // atan2(0,0) = 0
    float p = atan_poly01(t);
    p = (ay > ax)   ? (1.5707963267948966f - p) : p;
    p = (x < 0.0f)  ? (3.1415926535897932f - p) : p;
    return copysignf(p, y);
}

// sin & cos via Cody-Waite 3-term pi/2 reduction (exact for |x| << 1e3)
__device__ __forceinline__ void fast_sincosf(float x, float* sp_out, float* cp_out) {
    const float fn = rintf(x * 0.63661977236758134f);   // x * 2/pi
    const int   n  = (int)fn;
    float r = fmaf(fn, -1.5707962512969971e+00f, x);
    r = fmaf(fn, -7.5497894158615964e-08f, r);
    r = fmaf(fn, -5.3903029534742385e-15f, r);
    const float s = r * r;
    float sp = fmaf(s,  2.7557314297e-06f, -1.9841270114e-04f);
    sp = fmaf(s, sp,  8.3333337680e-03f);
    sp = fmaf(s, sp, -1.6666667163e-01f);
    const float sr = fmaf(r * s, sp, r);
    float cp = fmaf(s,  2.4801587642e-05f, -1.3888889225e-03f);
    cp = fmaf(s, cp,  4.1666667908e-02f);
    cp = fmaf(s, cp, -5.0000000000e-01f);
    const float cr = fmaf(s, cp, 1.0f);
    const bool swap = (n & 1) != 0;
    float so = swap ? cr : sr;
    float co = swap ? sr : cr;
    so = (n & 2)       ? -so : so;
    co = ((n + 1) & 2) ? -co : co;
    *sp_out = so;
    *cp_out = co;
}

// mlp(a) for a >= 0, 12 hidden units; dual accumulators halve the dot chain.
__device__ __forceinline__ float mlp12(float a,
                                       const float (&W1)[12],
                                       const float (&B1)[12],
                                       const float (&W2)[12],
                                       float b2) {
    float acc0 = b2, acc1 = 0.0f;
#pragma unroll
    for (int j = 0; j < 12; j += 2) {
        float h0 = fmaxf(fmaf(a, W1[j],     B1[j]),     0.0f);
        float h1 = fmaxf(fmaf(a, W1[j + 1], B1[j + 1]), 0.0f);
        acc0 = fmaf(h0, W2[j],     acc0);
        acc1 = fmaf(h1, W2[j + 1], acc1);
    }
    return acc0 + acc1;
}

struct __attribute__((aligned(4))) F3 { float x, y, z; };

__global__ __launch_bounds__(256) void dyn_model_kernel(
    const float* __restrict__ x,
    const float* __restrict__ rxw_p, const float* __restrict__ rxb_p,
    const float* __restrict__ fyW1,  const float* __restrict__ fyB1,
    const float* __restrict__ fyW2,  const float* __restrict__ fyB2,
    const float* __restrict__ ryW1,  const float* __restrict__ ryB1,
    const float* __restrict__ ryW2,  const float* __restrict__ ryB2,
    float* __restrict__ out, int n)
{
    // Stage the six 12-float weight vectors (288 B) into LDS via the gfx1250
    // async path, then broadcast-read them into registers once per thread.
    __shared__ __align__(16) float smem[72];
    const unsigned tid = threadIdx.x;
    if (tid < 3) {
        const unsigned c    = tid * 16;
        const unsigned base = (unsigned)(uintptr_t)(&smem[0]);
        async_b128_to_lds(base +   0 + c, (const char*)fyW1 + c);
        async_b128_to_lds(base +  48 + c, (const char*)fyB1 + c);
        async_b128_to_lds(base +  96 + c, (const char*)fyW2 + c);
        async_b128_to_lds(base + 144 + c, (const char*)ryW1 + c);
        async_b128_to_lds(base + 192 + c, (const char*)ryB1 + c);
        async_b128_to_lds(base + 240 + c, (const char*)ryW2 + c);
    }
    wait_async0();
    __syncthreads();

    float w1f[12], b1f[12], w2f[12], w1r[12], b1r[12], w2r[12];
#pragma unroll
    for (int j = 0; j < 12; ++j) {
        w1f[j] = smem[j];        b1f[j] = smem[12 + j];  w2f[j] = smem[24 + j];
        w1r[j] = smem[36 + j];   b1r[j] = smem[48 + j];  w2r[j] = smem[60 + j];
    }

    const float Rxw = rxw_p[0];
    const float Rxb = rxb_p[0];
    const float fb2 = fyB2[0];
    const float rb2 = ryB2[0];

    const float4* __restrict__ xv = reinterpret_cast<const float4*>(x);
    const int base_i = (int)(blockIdx.x * (256u * kEPT) + tid);

#pragma unroll 1   // keep rolled: amortize weight registers, small I$ footprint
    for (int e = 0; e < kEPT; ++e) {
        const int i = base_i + e * 256;
        if (i >= n) break;

        const float4 lo = xv[2 * i + 0];   // cols 0..3
        const float4 hi = xv[2 * i + 1];   // cols 4..7
        const float pwm   = lo.x;
        const float theta = lo.z;
        const float vx0   = lo.w;
        const float vy0   = hi.x;
        const float w0    = hi.y;
        const float pitch = hi.w;

        float sT, cT, sP, cP;
        fast_sincosf(theta, &sT, &cT);
        fast_sincosf(pitch, &sP, &cP);
        const float gsinP    = kG * sP;
        const float pwm_term = (pwm > 0.0f ? kCM1 : kCM2) * pwm * kInvMass;

        float o0 = 0.0f, o1 = 0.0f, o2 = 0.0f;
#pragma unroll
        for (int it = 0; it < 2; ++it) {
            const float vx = vx0 + o0;
            const float vy = vy0 + o1;
            const float w  = w0  + o2;

            const float af = theta - fast_atan2f(fmaf(w, kLF, vy), vx);
            const float ar = fast_atan2f(fmaf(w, kLR, -vy), vx);

            const float mf  = mlp12(fabsf(af), w1f, b1f, w2f, fb2);
            const float Ffy = (af > 0.0f) ? mf : -mf;
            const float mr  = mlp12(fabsf(ar), w1r, b1r, w2r, rb2);
            const float Fry = (ar > 0.0f) ? mr : -mr;

            const float Frx = fmaf(Rxw * vx, vx, Rxb) * kInvMass + pwm_term;

            const float vx_dot = Frx - Ffy * sT + vy * w - gsinP;
            const float vy_dot = Fry + Ffy * cT - vx * w;
            const float w_dot  = kMassIz * fmaf(Ffy * kLF, cT, -(Fry * kLR));

            o0 = fmaf(vx_dot, kDT, o0);
            o1 = fmaf(vy_dot, kDT, o1);
            o2 = fmaf(w_dot,  kDT, o2);
        }

        reinterpret_cast<F3*>(out)[i] = F3{o0, o1, o2};
    }
}

extern "C" void kernel_launch(void* const* d_in, const int* in_sizes, int n_in,
                              void* d_out, int out_size, void* d_ws, size_t ws_size,
                              hipStream_t stream) {
    (void)n_in; (void)out_size; (void)d_ws; (void)ws_size;
    const float* x    = (const float*)d_in[0];
    const float* rxw  = (const float*)d_in[1];
    const float* rxb  = (const float*)d_in[2];
    const float* fyW1 = (const float*)d_in[3];
    const float* fyB1 = (const float*)d_in[4];
    const float* fyW2 = (const float*)d_in[5];
    const float* fyB2 = (const float*)d_in[6];
    const float* ryW1 = (const float*)d_in[7];
    const float* ryB1 = (const float*)d_in[8];
    const float* ryW2 = (const float*)d_in[9];
    const float* ryB2 = (const float*)d_in[10];
    float* out = (float*)d_out;

    const int n = in_sizes[0] / 8;                       // B rows of 8 floats
    const int per_block = 256 * kEPT;
    const int blocks = (n + per_block - 1) / per_block;
    dyn_model_kernel<<<blocks, 256, 0, stream>>>(
        x, rxw, rxb, fyW1, fyB1, fyW2, fyB2, ryW1, ryB1, ryW2, ryB2, out, n);
}